// QCL_85280870629384
// MI455X (gfx1250) — compile-verified
//
#include <hip/hip_runtime.h>

#define NQ 14
#define NSTATE (1 << NQ)        // 16384 amplitudes
#define NPAIR  (NSTATE >> 1)    // 8192 gate pairs
#define TPB    256
#define QDEPTH 6
#define OUTD   10
#define BATCH  512

// ---------------------------------------------------------------------------
// Gate primitives operating on an LDS-resident statevector (split re/im).
// Each gate ends with a workgroup barrier; the next gate may repartition work.
// ---------------------------------------------------------------------------

__device__ __forceinline__ void ry_gate(float* re, float* im, int q, float c, float s) {
    const int tid = threadIdx.x;
    const int sh = (NQ - 1) - q;      // bit position of qubit q
    const int strd = 1 << sh;
    #pragma unroll 4
    for (int p = tid; p < NPAIR; p += TPB) {
        const int lo = p & (strd - 1);
        const int hi = p >> sh;
        const int i0 = (hi << (sh + 1)) | lo;
        const int i1 = i0 | strd;
        const float r0 = re[i0], r1 = re[i1];
        const float m0 = im[i0], m1 = im[i1];
        re[i0] = c * r0 - s * r1;
        re[i1] = s * r0 + c * r1;
        im[i0] = c * m0 - s * m1;
        im[i1] = s * m0 + c * m1;
    }
    __syncthreads();
}

__device__ __forceinline__ void rz_gate(float* re, float* im, int q, float c, float s) {
    const int tid = threadIdx.x;
    const int sh = (NQ - 1) - q;
    #pragma unroll 4
    for (int i = tid; i < NSTATE; i += TPB) {
        // bit==0: multiply by exp(-i*theta/2); bit==1: exp(+i*theta/2)
        const float sg = ((i >> sh) & 1) ? s : -s;
        const float r = re[i], m = im[i];
        re[i] = r * c - m * sg;
        im[i] = m * c + r * sg;
    }
    __syncthreads();
}

// CNOT with control=c, target=c+1 (adjacent bits: control at 13-c, target at 12-c)
__device__ __forceinline__ void cnot_adj(float* re, float* im, int c) {
    const int tid = threadIdx.x;
    const int bt = (NQ - 2) - c;              // target bit position
    #pragma unroll 4
    for (int p = tid; p < (NSTATE >> 2); p += TPB) {
        const int lo = p & ((1 << bt) - 1);
        const int hi = p >> bt;
        const int i0 = (hi << (bt + 2)) | (1 << (bt + 1)) | lo;  // ctrl=1, tgt=0
        const int i1 = i0 | (1 << bt);                            // ctrl=1, tgt=1
        float t;
        t = re[i0]; re[i0] = re[i1]; re[i1] = t;
        t = im[i0]; im[i0] = im[i1]; im[i1] = t;
    }
    __syncthreads();
}

// ---------------------------------------------------------------------------
// Kernel 1: one workgroup = one batch element; full 128 KB statevector in LDS.
// Produces Z as a zero-padded [BATCH][16] f32 matrix in workspace.
// ---------------------------------------------------------------------------
__global__ void __launch_bounds__(TPB)
qsim_kernel(const float* __restrict__ inputs,   // [BATCH][NQ]
            const float* __restrict__ weights,  // [QDEPTH][NQ]
            float* __restrict__ Z) {            // [BATCH][16]
    __shared__ float re[NSTATE];
    __shared__ float im[NSTATE];
    const int tid = threadIdx.x;
    const int b = blockIdx.x;

    for (int i = tid; i < NSTATE; i += TPB) {
        re[i] = (i == 0) ? 1.0f : 0.0f;
        im[i] = 0.0f;
    }
    __syncthreads();

    // Angle-encoding layer: RY(asin(x)) then RZ(acos(x^2)) per qubit
    for (int q = 0; q < NQ; ++q) {
        const float x = inputs[b * NQ + q];
        const float th = 0.5f * asinf(x);
        ry_gate(re, im, q, cosf(th), sinf(th));
        const float ph = 0.5f * acosf(x * x);
        rz_gate(re, im, q, cosf(ph), sinf(ph));
    }

    // Entangling layers: CNOT chain + RY(weights)
    for (int k = 0; k < QDEPTH; ++k) {
        for (int c = 0; c < NQ - 1; ++c)
            cnot_adj(re, im, c);
        for (int q = 0; q < NQ; ++q) {
            const float th = 0.5f * weights[k * NQ + q];
            ry_gate(re, im, q, cosf(th), sinf(th));
        }
    }

    // <Z_w> for all 14 qubits: per-thread partials, then deterministic tree reduce
    float acc[NQ];
    #pragma unroll
    for (int w = 0; w < NQ; ++w) acc[w] = 0.0f;
    for (int i = tid; i < NSTATE; i += TPB) {
        const float pr = re[i] * re[i] + im[i] * im[i];
        #pragma unroll
        for (int w = 0; w < NQ; ++w)
            acc[w] += ((i >> ((NQ - 1) - w)) & 1) ? -pr : pr;
    }
    __syncthreads();                    // done reading state; reuse re[] as scratch
    #pragma unroll
    for (int w = 0; w < NQ; ++w) re[w * TPB + tid] = acc[w];
    __syncthreads();
    for (int off = TPB / 2; off >= 1; off >>= 1) {
        if (tid < off) {
            #pragma unroll
            for (int w = 0; w < NQ; ++w)
                re[w * TPB + tid] += re[w * TPB + tid + off];
        }
        __syncthreads();
    }
    if (tid < 16)
        Z[b * 16 + tid] = (tid < NQ) ? re[tid * TPB] : 0.0f;   // zero-pad K to 16
}

// ---------------------------------------------------------------------------
// Kernel 2: out = Z(512x16, last 2 cols zero) @ W(14x10 padded) + bias
// via chained V_WMMA_F32_16X16X4_F32 (K = 4x4 = 16). One wave per 16x16 tile.
// ---------------------------------------------------------------------------
typedef float v2f __attribute__((ext_vector_type(2)));
typedef float v8f __attribute__((ext_vector_type(8)));

__global__ void __launch_bounds__(32)
dense_wmma_kernel(const float* __restrict__ Z,     // [BATCH][16]
                  const float* __restrict__ W,     // [NQ][OUTD]
                  const float* __restrict__ bias,  // [OUTD]
                  float* __restrict__ out) {       // [BATCH][OUTD]
    const int lane = threadIdx.x;          // 0..31, one full wave
    const int mtile = blockIdx.x;          // 0..31 (M = 512 / 16)
    const int half = lane >> 4;            // K/M half-select per ISA layout
    const int r = lane & 15;

    v8f c = {};
    #pragma unroll
    for (int s = 0; s < 4; ++s) {
        // A fragment (16x4): lanes 0-15 carry K = {4s, 4s+1}; lanes 16-31 K = {4s+2, 4s+3}
        const int k0 = 4 * s + (half ? 2 : 0);
        const int k1 = k0 + 1;
        v2f a, bf;
        a[0] = Z[(mtile * 16 + r) * 16 + k0];
        a[1] = Z[(mtile * 16 + r) * 16 + k1];
        // B fragment (4x16): same K striping, N = r across lanes; pad K>=14 / N>=10 with 0
        bf[0] = (k0 < NQ && r < OUTD) ? W[k0 * OUTD + r] : 0.0f;
        bf[1] = (k1 < NQ && r < OUTD) ? W[k1 * OUTD + r] : 0.0f;
        c = __builtin_amdgcn_wmma_f32_16x16x4_f32(
                /*neg_a=*/false, a, /*neg_b=*/false, bf,
                /*c_mod=*/(short)0, c, /*reuse_a=*/false, /*reuse_b=*/false);
    }

    // D layout: vgpr j = row (half*8 + j), col = r
    const int col = r;
    #pragma unroll
    for (int j = 0; j < 8; ++j) {
        const int row = mtile * 16 + half * 8 + j;
        if (col < OUTD)
            out[row * OUTD + col] = c[j] + bias[col];
    }
}

// ---------------------------------------------------------------------------
extern "C" void kernel_launch(void* const* d_in, const int* in_sizes, int n_in,
                              void* d_out, int out_size, void* d_ws, size_t ws_size,
                              hipStream_t stream) {
    const float* inputs  = (const float*)d_in[0];  // [512][14]
    const float* weights = (const float*)d_in[1];  // [6][14]
    const float* dense_w = (const float*)d_in[2];  // [14][10]
    const float* dense_b = (const float*)d_in[3];  // [10]
    float* out = (float*)d_out;                    // [512][10]
    float* Z = (float*)d_ws;                       // [512][16] scratch

    qsim_kernel<<<BATCH, TPB, 0, stream>>>(inputs, weights, Z);
    dense_wmma_kernel<<<BATCH / 16, 32, 0, stream>>>(Z, dense_w, dense_b, out);
}